// CircuitGNN_83571473645755
// MI455X (gfx1250) — compile-verified
//
#include <hip/hip_runtime.h>
#include <hip/hip_bf16.h>

typedef float v2f __attribute__((ext_vector_type(2)));
typedef float v8f __attribute__((ext_vector_type(8)));

#define NEG_SLOPE 0.2f
#define SORT_NEG_INF 0x007FFFFFu   // f2s(-inf)

// ---- helpers -------------------------------------------------------------
__device__ __forceinline__ unsigned f2s(float f) {
    unsigned u = __float_as_uint(f);
    return (u & 0x80000000u) ? ~u : (u | 0x80000000u);
}
__device__ __forceinline__ float s2f(unsigned u) {
    return (u & 0x80000000u) ? __uint_as_float(u & 0x7fffffffu)
                             : __uint_as_float(~u);
}
__device__ __forceinline__ void edge_sd(const int* __restrict__ ei,
                                        long long E, long long e,
                                        int& s, int& d) {
    if (e < E) { s = ei[e]; d = ei[E + e]; }
    else       { s = d = (int)(e - E); }
}

// ---- generic fill --------------------------------------------------------
__global__ void fill_u32(unsigned* __restrict__ p, unsigned v, long long n) {
    long long t = (long long)blockIdx.x * blockDim.x + threadIdx.x;
    if (t < n) p[t] = v;
}

// ---- layer-1 GEMM: h1[N,128] = x[N,4] @ W1[128,4]^T  (K=4 -> 1 WMMA/tile)
__global__ void gemm1_wmma(const float* __restrict__ x,
                           const float* __restrict__ w1,
                           float* __restrict__ h1, int n) {
    int wave = (int)((blockIdx.x * blockDim.x + threadIdx.x) >> 5);
    int lane = threadIdx.x & 31;
    int m0 = wave * 16;                 // node tile base (wave-uniform)
    if (m0 >= n) return;                // uniform exit: EXEC all-1 for WMMA
    int kh = lane >> 4;                 // 0: K=0,1   1: K=2,3
    int m  = m0 + (lane & 15);
    int mc = m < n ? m : n - 1;         // clamped load, store handled below
    v2f a;
    a.x = x[mc * 4 + kh * 2 + 0];
    a.y = x[mc * 4 + kh * 2 + 1];
    bool full = (m0 + 16 <= n);         // wave-uniform fast path
    int rbase = m0 + (kh << 3);
    for (int t = 0; t < 8; ++t) {       // 8 col tiles -> 128 outputs
        int col = t * 16 + (lane & 15);
        v2f b;
        b.x = w1[col * 4 + kh * 2 + 0];
        b.y = w1[col * 4 + kh * 2 + 1];
        v8f c = {};
        c = __builtin_amdgcn_wmma_f32_16x16x4_f32(false, a, false, b,
                                                  (short)0, c, false, false);
        float* outp = h1 + (size_t)rbase * 128 + col;
        if (full) {
#pragma unroll
            for (int i = 0; i < 8; ++i) outp[(size_t)i * 128] = c[i];
        } else {
#pragma unroll
            for (int i = 0; i < 8; ++i)
                if (rbase + i < n) outp[(size_t)i * 128] = c[i];
        }
    }
}

// ---- layer-2 GEMM: h2[N,64] = x2[N,128] @ W2[64,128]^T, W2 async-staged in LDS
__global__ void gemm2_wmma(const float* __restrict__ x2,
                           const float* __restrict__ w2,
                           float* __restrict__ h2, int n) {
    __shared__ float lw[64 * 132];      // pad 128 -> 132 (bank-conflict free)
    // async copy W2 -> LDS: 2048 16-byte chunks, ASYNCcnt-tracked
    for (int c = threadIdx.x; c < 64 * 32; c += blockDim.x) {
        int row = c >> 5;
        int kc  = (c & 31) << 2;                       // float index, mult of 4
        unsigned ldsa = (unsigned)(size_t)&lw[row * 132 + kc];
        const float* g = w2 + row * 128 + kc;
        asm volatile("global_load_async_to_lds_b128 %0, %1, off"
                     :: "v"(ldsa), "v"(g) : "memory");
    }
    asm volatile("s_wait_asynccnt 0x0" ::: "memory");
    __syncthreads();

    int wave = threadIdx.x >> 5;        // 8 waves/block
    int lane = threadIdx.x & 31;
    int nodeTile = blockIdx.x * 2 + (wave >> 2);
    int colTile  = wave & 3;
    int m0 = nodeTile * 16;
    if (m0 >= n) return;                // wave-uniform
    int kh = lane >> 4;
    int m  = m0 + (lane & 15);
    int mc = m < n ? m : n - 1;
    int ncol = colTile * 16 + (lane & 15);
    const float* __restrict__ arow = x2 + (size_t)mc * 128;
    const float* brow = lw + ncol * 132;
    v8f c = {};
#pragma unroll
    for (int kk = 0; kk < 128; kk += 4) {
        v2f a, b;
        a.x = arow[kk + kh * 2 + 0];
        a.y = arow[kk + kh * 2 + 1];
        b.x = brow[kk + kh * 2 + 0];
        b.y = brow[kk + kh * 2 + 1];
        c = __builtin_amdgcn_wmma_f32_16x16x4_f32(false, a, false, b,
                                                  (short)0, c, false, false);
    }
    int rbase = m0 + (kh << 3);
    float* outp = h2 + (size_t)rbase * 64 + colTile * 16 + (lane & 15);
    if (m0 + 16 <= n) {                 // wave-uniform fast path
#pragma unroll
        for (int i = 0; i < 8; ++i) outp[(size_t)i * 64] = c[i];
    } else {
#pragma unroll
        for (int i = 0; i < 8; ++i)
            if (rbase + i < n) outp[(size_t)i * 64] = c[i];
    }
}

// ---- attention logits: as[n,h]=dot(h_row, att_src[h]), ad likewise --------
template <int H, int D>
__global__ void alpha_dots(const float* __restrict__ h,
                           const float* __restrict__ att_s,
                           const float* __restrict__ att_d,
                           float* __restrict__ as_, float* __restrict__ ad_,
                           int n) {
    int t = blockIdx.x * blockDim.x + threadIdx.x;
    if (t >= n * H) return;
    int node = t / H, hd = t % H;       // H compile-time -> shifts
    const float* row = h + (size_t)node * (H * D) + hd * D;
    const float* sv = att_s + hd * D;
    const float* dv = att_d + hd * D;
    float accs = 0.f, accd = 0.f;
#pragma unroll 8
    for (int i = 0; i < D; ++i) {
        float v = row[i];
        accs += v * sv[i];
        accd += v * dv[i];
    }
    as_[t] = accs;
    ad_[t] = accd;
}

// ---- edge pass 1: e = leaky_relu(as[src]+ad[dst]); segment max ------------
template <int H>
__global__ void epass1(const int* __restrict__ ei, long long E, long long Et,
                       const float* __restrict__ as_,
                       const float* __restrict__ ad_,
                       float* __restrict__ ew, unsigned* __restrict__ emax) {
    long long t = (long long)blockIdx.x * blockDim.x + threadIdx.x;
    if (t >= Et * H) return;
    long long e = t / H;                // compile-time H -> shift
    int hd = (int)(t % H);
    int s, d; edge_sd(ei, E, e, s, d);
    float v = as_[(size_t)s * H + hd] + ad_[(size_t)d * H + hd];
    v = v > 0.f ? v : NEG_SLOPE * v;
    ew[t] = v;
    atomicMax(&emax[(size_t)d * H + hd], f2s(v));
}

// ---- edge pass 2: w = exp(e - max); segment sum ---------------------------
template <int H>
__global__ void epass2(const int* __restrict__ ei, long long E, long long Et,
                       float* __restrict__ ew,
                       const unsigned* __restrict__ emax,
                       float* __restrict__ denom) {
    long long t = (long long)blockIdx.x * blockDim.x + threadIdx.x;
    if (t >= Et * H) return;
    long long e = t / H;
    int hd = (int)(t % H);
    int s, d; edge_sd(ei, E, e, s, d);
    (void)s;
    float w = __expf(ew[t] - s2f(emax[(size_t)d * H + hd]));
    ew[t] = w;
    atomicAdd(&denom[(size_t)d * H + hd], w);
}

// ---- edge pass 3, layer 1: 128 feats, wave/edge, lane owns a float4 -------
__global__ void eagg128(const int* __restrict__ ei, long long E, long long Et,
                        const float* __restrict__ ew,
                        const float* __restrict__ denom,
                        const float* __restrict__ h, float* __restrict__ agg) {
    long long t = (long long)blockIdx.x * blockDim.x + threadIdx.x;
    long long e = t >> 5;
    if (e >= Et) return;
    int lane = (int)(t & 31);
    int s, d; edge_sd(ei, E, e, s, d);
    int off = lane << 2;            // 0..124
    int hd = off >> 6;              // head 0 or 1
    float alpha = ew[e * 2 + hd] / denom[(size_t)d * 2 + hd];
    const float4 hv = *(const float4*)(h + (size_t)s * 128 + off);
    float* ag = agg + (size_t)d * 128 + off;
    atomicAdd(ag + 0, alpha * hv.x);
    atomicAdd(ag + 1, alpha * hv.y);
    atomicAdd(ag + 2, alpha * hv.z);
    atomicAdd(ag + 3, alpha * hv.w);
}

// ---- edge pass 3, layer 2: 64 feats, wave/edge, lane owns a float2 --------
__global__ void eagg64(const int* __restrict__ ei, long long E, long long Et,
                       const float* __restrict__ ew,
                       const float* __restrict__ denom,
                       const float* __restrict__ h, float* __restrict__ agg) {
    long long t = (long long)blockIdx.x * blockDim.x + threadIdx.x;
    long long e = t >> 5;
    if (e >= Et) return;
    int lane = (int)(t & 31);
    int s, d; edge_sd(ei, E, e, s, d);
    int off = lane << 1;            // 0..62
    float alpha = ew[e] / denom[d];
    const float2 hv = *(const float2*)(h + (size_t)s * 64 + off);
    float* ag = agg + (size_t)d * 64 + off;
    atomicAdd(ag + 0, alpha * hv.x);
    atomicAdd(ag + 1, alpha * hv.y);
}

// ---- bias + ELU in place (F power of two, pass mask) ----------------------
__global__ void bias_elu(float* __restrict__ a, const float* __restrict__ b,
                         long long total, int fmask) {
    long long t = (long long)blockIdx.x * blockDim.x + threadIdx.x;
    if (t >= total) return;
    float v = a[t] + b[(int)(t & fmask)];
    a[t] = v > 0.f ? v : (__expf(v) - 1.f);
}

// ---- layer-2 finish: bias+ELU, accumulate graph sums ----------------------
__global__ void finish_pool(float* __restrict__ a, const float* __restrict__ b,
                            const int* __restrict__ batch,
                            float* __restrict__ psum, long long n64) {
    long long t = (long long)blockIdx.x * blockDim.x + threadIdx.x;
    if (t >= n64) return;
    int node = (int)(t >> 6), dcol = (int)(t & 63);
    float v = a[t] + b[dcol];
    v = v > 0.f ? v : (__expf(v) - 1.f);
    a[t] = v;
    atomicAdd(&psum[batch[node] * 64 + dcol], v);
}

__global__ void count_nodes(const int* __restrict__ batch,
                            float* __restrict__ pcnt, int n) {
    int t = blockIdx.x * blockDim.x + threadIdx.x;
    if (t >= n) return;
    atomicAdd(&pcnt[batch[t]], 1.f);
}

// ---- readout head: out[g,s] = mean_pool(g) . wp[s] + bp[s] ----------------
__global__ void head_out(const float* __restrict__ psum,
                         const float* __restrict__ pcnt,
                         const float* __restrict__ wp,
                         const float* __restrict__ bp,
                         float* __restrict__ out) {
    int t = blockIdx.x * blockDim.x + threadIdx.x;
    if (t >= 64 * 16) return;
    int g = t >> 4, s = t & 15;
    float inv = 1.f / fmaxf(pcnt[g], 1.f);
    float acc = bp[s];
#pragma unroll 8
    for (int d = 0; d < 64; ++d)
        acc += psum[g * 64 + d] * inv * wp[s * 64 + d];
    out[t] = acc;
}

// ---------------------------------------------------------------------------
extern "C" void kernel_launch(void* const* d_in, const int* in_sizes, int n_in,
                              void* d_out, int out_size, void* d_ws,
                              size_t ws_size, hipStream_t stream) {
    const float* x        = (const float*)d_in[0];
    const int*   ei       = (const int*)d_in[1];
    const int*   batch    = (const int*)d_in[2];
    const float* w1       = (const float*)d_in[3];
    const float* att_src1 = (const float*)d_in[4];
    const float* att_dst1 = (const float*)d_in[5];
    const float* b1       = (const float*)d_in[6];
    const float* w2       = (const float*)d_in[7];
    const float* att_src2 = (const float*)d_in[8];
    const float* att_dst2 = (const float*)d_in[9];
    const float* b2       = (const float*)d_in[10];
    const float* wp       = (const float*)d_in[11];
    const float* bp       = (const float*)d_in[12];
    float* out = (float*)d_out;

    const int N = in_sizes[0] / 4;
    const long long E  = in_sizes[1] / 2;
    const long long Et = E + N;

    // workspace carve-out (16B aligned)
    char* p = (char*)d_ws;
    auto carve = [&](size_t bytes) {
        void* r = (void*)p;
        p += (bytes + 15) & ~(size_t)15;
        return r;
    };
    float*    h1    = (float*)carve((size_t)N * 128 * 4);  // also h2 later
    float*    agg1  = (float*)carve((size_t)N * 128 * 4);  // becomes x2
    float*    agg2  = (float*)carve((size_t)N * 64 * 4);
    float*    as_   = (float*)carve((size_t)N * 2 * 4);
    float*    ad_   = (float*)carve((size_t)N * 2 * 4);
    unsigned* emax  = (unsigned*)carve((size_t)N * 2 * 4);
    float*    denom = (float*)carve((size_t)N * 2 * 4);
    float*    ew    = (float*)carve((size_t)Et * 2 * 4);
    float*    psum  = (float*)carve((size_t)(64 * 64 + 64) * 4); // psum|pcnt
    float*    pcnt  = psum + 64 * 64;
    float*    h2    = h1;                                  // reuse after pass 3

    const int B = 256;
    auto cdiv = [](long long a, long long b) { return (int)((a + b - 1) / b); };

    // init accumulators (every call; graph-replay safe)
    fill_u32<<<cdiv((long long)N * 2, B), B, 0, stream>>>(emax, SORT_NEG_INF, (long long)N * 2);
    fill_u32<<<cdiv((long long)N * 2, B), B, 0, stream>>>((unsigned*)denom, 0u, (long long)N * 2);
    fill_u32<<<cdiv((long long)N * 128, B), B, 0, stream>>>((unsigned*)agg1, 0u, (long long)N * 128);
    fill_u32<<<cdiv((long long)N * 64, B), B, 0, stream>>>((unsigned*)agg2, 0u, (long long)N * 64);
    fill_u32<<<cdiv(64 * 64 + 64, B), B, 0, stream>>>((unsigned*)psum, 0u, 64 * 64 + 64);

    // ---- layer 1 (H=2, D=64, concat) ----
    {
        int waves = cdiv(N, 16);
        gemm1_wmma<<<cdiv(waves, 8), B, 0, stream>>>(x, w1, h1, N);
    }
    alpha_dots<2, 64><<<cdiv((long long)N * 2, B), B, 0, stream>>>(h1, att_src1, att_dst1, as_, ad_, N);
    epass1<2><<<cdiv(Et * 2, B), B, 0, stream>>>(ei, E, Et, as_, ad_, ew, emax);
    epass2<2><<<cdiv(Et * 2, B), B, 0, stream>>>(ei, E, Et, ew, emax, denom);
    eagg128<<<cdiv(Et * 32, B), B, 0, stream>>>(ei, E, Et, ew, denom, h1, agg1);
    bias_elu<<<cdiv((long long)N * 128, B), B, 0, stream>>>(agg1, b1, (long long)N * 128, 127);

    // reset per-dst state for layer 2 (H=1)
    fill_u32<<<cdiv((long long)N, B), B, 0, stream>>>(emax, SORT_NEG_INF, (long long)N);
    fill_u32<<<cdiv((long long)N, B), B, 0, stream>>>((unsigned*)denom, 0u, (long long)N);

    // ---- layer 2 (H=1, D=64, mean==identity) ----
    {
        int nodeTiles = cdiv(N, 16);
        gemm2_wmma<<<cdiv(nodeTiles, 2), B, 0, stream>>>(agg1, w2, h2, N);
    }
    alpha_dots<1, 64><<<cdiv((long long)N, B), B, 0, stream>>>(h2, att_src2, att_dst2, as_, ad_, N);
    epass1<1><<<cdiv(Et, B), B, 0, stream>>>(ei, E, Et, as_, ad_, ew, emax);
    epass2<1><<<cdiv(Et, B), B, 0, stream>>>(ei, E, Et, ew, emax, denom);
    eagg64<<<cdiv(Et * 32, B), B, 0, stream>>>(ei, E, Et, ew, denom, h2, agg2);

    // ---- finish: bias+ELU, mean pool, head ----
    finish_pool<<<cdiv((long long)N * 64, B), B, 0, stream>>>(agg2, b2, batch, psum, (long long)N * 64);
    count_nodes<<<cdiv((long long)N, B), B, 0, stream>>>(batch, pcnt, N);
    head_out<<<cdiv(64 * 16, B), B, 0, stream>>>(psum, pcnt, wp, bp, out);
}